// GCNLayer_47665547051751
// MI455X (gfx1250) — compile-verified
//
#include <hip/hip_runtime.h>
#include <hip/hip_bf16.h>

typedef __attribute__((ext_vector_type(2))) float v2f;
typedef __attribute__((ext_vector_type(8))) float v8f;

#define N_NODES 50000
#define D_FEAT  64

// ---------------------------------------------------------------------------
// Kernel 1: zero the accumulator (d_out doubles as `support` then `output`).
// ---------------------------------------------------------------------------
__global__ void gcn_zero(float4* __restrict__ p, int n4) {
    int i = blockIdx.x * blockDim.x + threadIdx.x;
    if (i < n4) p[i] = make_float4(0.f, 0.f, 0.f, 0.f);
}

// ---------------------------------------------------------------------------
// Kernel 2: COO SpMM scatter:  support[dst] += features[src] * w
// 16 threads per edge, each thread owns a float4 feature chunk.
// Consecutive threads touch consecutive 16B chunks of one src row ->
// fully coalesced 256B gathers; scatter uses global_atomic_add_f32 (L2 RMW).
// ---------------------------------------------------------------------------
__global__ void gcn_scatter(const float* __restrict__ feat,
                            const int*   __restrict__ esrc,
                            const int*   __restrict__ edst,
                            const float* __restrict__ ew,
                            float*       __restrict__ support,
                            int nEdges) {
    int t = blockIdx.x * blockDim.x + threadIdx.x;
    int e = t >> 4;              // edge index (16 threads per edge)
    int c = (t & 15) << 2;       // feature chunk start (0,4,...,60)
    if (e >= nEdges) return;

    int   s = esrc[e];
    int   d = edst[e];
    float w = ew[e];

    float4 f = *(const float4*)(feat + s * D_FEAT + c);
    float* o = support + d * D_FEAT + c;
    atomicAdd(o + 0, f.x * w);
    atomicAdd(o + 1, f.y * w);
    atomicAdd(o + 2, f.z * w);
    atomicAdd(o + 3, f.w * w);
}

// ---------------------------------------------------------------------------
// Kernel 3: in-place tile GEMM via V_WMMA_F32_16X16X4_F32.
//   out[n, o] = sum_d support[n, d] * W[o, d]      (D = A x B, B = W^T)
// One wave (32 lanes) per 16-row tile; 4 N-tiles of 16 cols; K=64 in 16
// steps of 4. A fragments are fully loaded to VGPRs before the in-place
// store, so overwriting the tile is safe. EXEC is all-1s (no divergence).
//
// ISA layouts used (CDNA5 ISA §7.12.2):
//   A (16x4 f32):  lane L (m=L&15, hi=L>>4) holds A[m, hi*2 + {0,1}]
//   B (4x16 f32):  VGPR v, lane L holds B[v + hi*2, L&15]  (mirrors C/D)
//   C/D (16x16):   VGPR v, lane L holds D[v + hi*8, L&15]
// ---------------------------------------------------------------------------
__global__ __launch_bounds__(32) void gcn_gemm_inplace(float* __restrict__ out,
                                                       const float* __restrict__ W) {
    const int row0 = blockIdx.x * 16;
    const int lane = threadIdx.x;
    const int m    = lane & 15;
    const int hi   = lane >> 4;

    // Preload all A fragments for this lane: support[row0+m][ks*4 + hi*2 + {0,1}]
    v2f a[16];
    const float* arow = out + (row0 + m) * D_FEAT + hi * 2;
    #pragma unroll
    for (int ks = 0; ks < 16; ++ks) {
        a[ks] = *(const v2f*)(arow + ks * 4);
    }

    v8f acc[4];
    #pragma unroll
    for (int nt = 0; nt < 4; ++nt) {
        // B fragment source: B[k][n] = W[n][k]; per lane n = nt*16 + m,
        // k = ks*4 + hi*2 + {0,1}  -> contiguous float2 along W's row.
        const float* wrow = W + (nt * 16 + m) * D_FEAT + hi * 2;
        v8f c = {};
        #pragma unroll
        for (int ks = 0; ks < 16; ++ks) {
            v2f b = *(const v2f*)(wrow + ks * 4);
            c = __builtin_amdgcn_wmma_f32_16x16x4_f32(
                    /*neg_a=*/false, a[ks],
                    /*neg_b=*/false, b,
                    /*c_mod=*/(short)0, c,
                    /*reuse_a=*/false, /*reuse_b=*/false);
        }
        acc[nt] = c;
    }

    // Write D over the same 16x64 tile (all A reads already in registers).
    #pragma unroll
    for (int nt = 0; nt < 4; ++nt) {
        #pragma unroll
        for (int v = 0; v < 8; ++v) {
            out[(row0 + v + hi * 8) * D_FEAT + nt * 16 + m] = acc[nt][v];
        }
    }
}

// ---------------------------------------------------------------------------
extern "C" void kernel_launch(void* const* d_in, const int* in_sizes, int n_in,
                              void* d_out, int out_size, void* d_ws, size_t ws_size,
                              hipStream_t stream) {
    const float* feat = (const float*)d_in[0];   // [50000, 64] f32
    const int*   esrc = (const int*)  d_in[1];   // [800000] i32
    const int*   edst = (const int*)  d_in[2];   // [800000] i32
    const float* ew   = (const float*)d_in[3];   // [800000] f32
    const float* W    = (const float*)d_in[4];   // [64, 64] f32
    float*       out  = (float*)d_out;           // [50000, 64] f32

    const int nEdges = in_sizes[1];

    // 1) zero accumulator
    {
        int n4 = (N_NODES * D_FEAT) / 4;
        int blocks = (n4 + 255) / 256;
        gcn_zero<<<blocks, 256, 0, stream>>>((float4*)out, n4);
    }
    // 2) gather * w, scatter-add
    {
        long long total = (long long)nEdges * 16;
        int blocks = (int)((total + 255) / 256);
        gcn_scatter<<<blocks, 256, 0, stream>>>(feat, esrc, edst, ew, out, nEdges);
    }
    // 3) in-place GEMM with f32 WMMA (50000 % 16 == 0 -> 3125 full tiles)
    {
        int blocks = N_NODES / 16;
        gcn_gemm_inplace<<<blocks, 32, 0, stream>>>(out, W);
    }
}